// TopKActivationFn_77008763617645
// MI455X (gfx1250) — compile-verified
//
#include <hip/hip_runtime.h>

typedef unsigned int u32;
typedef float f32x4 __attribute__((ext_vector_type(4)));
typedef u32   u32x4 __attribute__((ext_vector_type(4)));
typedef int   i32x4 __attribute__((ext_vector_type(4)));

// typed address-space pointers for the async-DMA builtin
typedef __attribute__((address_space(1))) i32x4 gas_i32x4;
typedef __attribute__((address_space(3))) i32x4 las_i32x4;

#define TOPK_D 32768
#define KSEL   64
#define NT     512                    // 16 wave32 per block
#define V4PT   (TOPK_D / (4 * NT))    // 16 float4 (64 floats) per thread
#define HCOPY  16                     // replicated LDS histogram copies

static_assert(V4PT == 16, "layout assumption");

// ---- CDNA5 async global->LDS path (guarded; fallback = plain b128 loads) ----
#if defined(__has_builtin)
#  if __has_builtin(__builtin_amdgcn_global_load_async_to_lds_b128)
#    define USE_ASYNC 1
#  endif
#endif
#ifndef USE_ASYNC
#  define USE_ASYNC 0
#endif

#if USE_ASYNC
#  if __has_builtin(__builtin_amdgcn_s_wait_asynccnt)
#    define ASYNC_WAIT(n) __builtin_amdgcn_s_wait_asynccnt(n)
#  else
#    define ASYNC_WAIT(n) asm volatile("s_wait_asynccnt " #n ::: "memory")
#  endif
#endif

// float -> monotonic key (larger key <=> larger float)
__device__ __forceinline__ u32 mono1(u32 u) {
  return (u & 0x80000000u) ? ~u : (u | 0x80000000u);
}

// block-wide exclusive scan over NT=512 values (wave32 shuffle scan + wave offsets)
__device__ __forceinline__ u32 block_exscan(u32 v, u32* s_ws) {
  const int lane = threadIdx.x & 31;
  const int wave = threadIdx.x >> 5;
  u32 x = v;
#pragma unroll
  for (int d = 1; d < 32; d <<= 1) {
    u32 y = __shfl_up(x, d, 32);
    if (lane >= d) x += y;
  }
  __syncthreads();                       // protect s_ws reuse from previous call
  if (lane == 31) s_ws[wave] = x;        // per-wave inclusive total
  __syncthreads();
  if (wave == 0) {
    u32 w  = (lane < (NT / 32)) ? s_ws[lane] : 0u;
    u32 xi = w;
#pragma unroll
    for (int d = 1; d < 32; d <<= 1) {
      u32 y = __shfl_up(xi, d, 32);
      if (lane >= d) xi += y;
    }
    if (lane < (NT / 32)) s_ws[lane] = xi - w;   // exclusive wave offsets
  }
  __syncthreads();
  return x - v + s_ws[wave];
}

__global__ __launch_bounds__(NT) void topk_relu_scatter(
    const float* __restrict__ x, float* __restrict__ out,
    float* __restrict__ idx_out)
{
  __shared__ u32 s_hist[256 * HCOPY];    // 16 KB replicated histogram
  __shared__ u32 s_ws[NT / 32];
  __shared__ u32 s_prefix, s_remain;
#if USE_ASYNC
  __shared__ float s_buf[2][4 * NT * 2]; // 2 x 16 KB staging (double buffer)
#endif

  const int tid = threadIdx.x;
  const int row = blockIdx.x;
  const float* __restrict__ src = x + (size_t)row * TOPK_D;

  u32x4 kv[V4PT];                        // row resident in VGPRs as keys

#if USE_ASYNC
  // Per-wave async DMA pipeline. Each thread reads back exactly the LDS
  // bytes its own lanes DMA'd, so ASYNCcnt waits are sufficient (no barrier).
  auto issue_chunk = [&](int c) {
    float* b = s_buf[c & 1];
#pragma unroll
    for (int h = 0; h < 2; ++h) {
      const float* g = src + ((2 * c + h) * NT + tid) * 4;
      float*       l = b + (h * NT + tid) * 4;
      __builtin_amdgcn_global_load_async_to_lds_b128(
          (gas_i32x4*)g, (las_i32x4*)l, 0, 0);
    }
  };
  auto read_chunk = [&](int c) {
    const float* b = s_buf[c & 1];
#pragma unroll
    for (int h = 0; h < 2; ++h) {
      u32x4 raw = *(const u32x4*)(b + (h * NT + tid) * 4);
      u32x4 kk;
      kk.x = mono1(raw.x); kk.y = mono1(raw.y);
      kk.z = mono1(raw.z); kk.w = mono1(raw.w);
      kv[2 * c + h] = kk;
    }
    // make sure the ds reads drained before this buffer is DMA-overwritten
    asm volatile("s_wait_dscnt 0" ::: "memory");
  };

  issue_chunk(0);
  issue_chunk(1);
  ASYNC_WAIT(2); read_chunk(0); issue_chunk(2);
  ASYNC_WAIT(2); read_chunk(1); issue_chunk(3);
  ASYNC_WAIT(2); read_chunk(2); issue_chunk(4);
  ASYNC_WAIT(2); read_chunk(3); issue_chunk(5);
  ASYNC_WAIT(2); read_chunk(4); issue_chunk(6);
  ASYNC_WAIT(2); read_chunk(5); issue_chunk(7);
  ASYNC_WAIT(2); read_chunk(6);
  ASYNC_WAIT(0); read_chunk(7);
#else
#pragma unroll
  for (int j = 0; j < V4PT; ++j) {
    u32x4 raw = *(const u32x4*)(src + (j * NT + tid) * 4);
    u32x4 kk;
    kk.x = mono1(raw.x); kk.y = mono1(raw.y);
    kk.z = mono1(raw.z); kk.w = mono1(raw.w);
    kv[j] = kk;
  }
#endif

  // ---- 4-pass MSB radix select for the KSEL-th largest key ----
  if (tid == 0) { s_prefix = 0u; s_remain = KSEL; }

  for (int p = 0; p < 4; ++p) {
    for (int i = tid; i < 256 * HCOPY; i += NT) s_hist[i] = 0u;
    __syncthreads();

    const int shift  = 24 - 8 * p;
    const u32 pref   = s_prefix;
    const u32 remain = s_remain;
    const u32 pmask  = (p == 0) ? 0u : (u32)(0xFFFFFFFFu << (shift + 8));
    const int hl     = tid & (HCOPY - 1);

#pragma unroll
    for (int j = 0; j < V4PT; ++j)
#pragma unroll
      for (int c = 0; c < 4; ++c) {
        u32 kk = kv[j][c];
        if ((kk & pmask) == pref)
          atomicAdd(&s_hist[(((kk >> shift) & 255u) << 4) + hl], 1u);
      }
    __syncthreads();

    // descending-bin suffix counts via block scan; unique winner picks the bin
    u32 binc = 0u;
    if (tid < 256) {
#pragma unroll
      for (int i = 0; i < HCOPY; ++i) binc += s_hist[((255 - tid) << 4) + i];
    }
    u32 excl = block_exscan(binc, s_ws);   // # elements in bins strictly above
    if (tid < 256 && excl < remain && excl + binc >= remain) {
      s_prefix = pref | ((u32)(255 - tid) << shift);
      s_remain = remain - excl;
    }
    // next pass's clear + barrier publishes s_prefix/s_remain
  }
  __syncthreads();
  const u32 T    = s_prefix;   // key of the KSEL-th largest element
  const u32 n_eq = s_remain;   // how many key==T elements to take

  // ---- deterministic compaction (ties broken in fixed thread/element order) ----
  u32 cnt_gt = 0, cnt_eq = 0;
#pragma unroll
  for (int j = 0; j < V4PT; ++j)
#pragma unroll
    for (int c = 0; c < 4; ++c) {
      u32 kk = kv[j][c];
      cnt_gt += (kk > T);
      cnt_eq += (kk == T);
    }
  const u32 eq_off = block_exscan(cnt_eq, s_ws);
  const u32 avail  = (eq_off < n_eq) ? (n_eq - eq_off) : 0u;
  const u32 my_eq  = (cnt_eq < avail) ? cnt_eq : avail;
  const u32 pos0   = block_exscan(cnt_gt + my_eq, s_ws);

  // ---- scatter: full-row streaming store (nontemporal) + idx append ----
  float* __restrict__ dst  = out + (size_t)row * TOPK_D;
  float* __restrict__ irow = idx_out + (size_t)row * KSEL;
  u32 pos = pos0, eq_seen = 0;
#pragma unroll
  for (int j = 0; j < V4PT; ++j) {
    const int e4 = j * NT + tid;
    f32x4 o;
#pragma unroll
    for (int c = 0; c < 4; ++c) {
      const u32 kk = kv[j][c];
      bool take = (kk > T);
      if (kk == T) { take = (eq_off + eq_seen) < n_eq; ++eq_seen; }
      // ReLU: only strictly-positive floats survive; key>0x80000000 <=> f>0
      o[c] = (take && kk > 0x80000000u)
                 ? __uint_as_float(kk ^ 0x80000000u) : 0.0f;
      if (take) { irow[pos] = (float)(e4 * 4 + c); ++pos; }
    }
    __builtin_nontemporal_store(o, (f32x4*)(dst + (size_t)e4 * 4));
  }
}

extern "C" void kernel_launch(void* const* d_in, const int* in_sizes, int n_in,
                              void* d_out, int out_size, void* d_ws, size_t ws_size,
                              hipStream_t stream) {
  const float* x = (const float*)d_in[0];
  const int B = in_sizes[0] / TOPK_D;          // 4096
  float* out = (float*)d_out;                  // [B, D] result
  float* idx = out + (size_t)B * TOPK_D;       // [B, K] indices (as float)
  topk_relu_scatter<<<dim3(B), dim3(NT), 0, stream>>>(x, out, idx);
}